// GIN_5497558139682
// MI455X (gfx1250) — compile-verified
//
#include <hip/hip_runtime.h>

// ---------------- problem constants (match reference) ----------------
#define N_GRAPHS 8192
#define NODES_PG 62
#define N_TOT (N_GRAPHS * NODES_PG)      // 507904
#define DEG 8
#define E_TOT (N_TOT * DEG)              // 4063232
#define HID 64
#define BN_EPS 1e-5f

typedef float v2f __attribute__((ext_vector_type(2)));
typedef float v8f __attribute__((ext_vector_type(8)));

__device__ __forceinline__ v8f splat8(float x) {
    v8f r;
#pragma unroll
    for (int i = 0; i < 8; ++i) r[i] = x;
    return r;
}

// ---------------- aggregation init / scatter ----------------

// AGG1[i] = x[i]  (N float4's, self term for layer 1)
__global__ __launch_bounds__(256) void init_copy4_kernel(
    const float4* __restrict__ src, float4* __restrict__ dst) {
    int id = blockIdx.x * 256 + threadIdx.x;   // N_TOT threads
    dst[id] = src[id];
}

// layer-1 scatter: AGG1[dst] += x[src]  (4 channels)
__global__ __launch_bounds__(256) void scatter4_kernel(
    const float* __restrict__ x, const int* __restrict__ src,
    const int* __restrict__ dst, float* __restrict__ agg) {
    int e = blockIdx.x * 256 + threadIdx.x;    // E_TOT threads
    int s = src[e], d = dst[e];
    float4 v = *(const float4*)(x + (size_t)s * 4);
    float* o = agg + (size_t)d * 4;
    atomicAdd(o + 0, v.x); atomicAdd(o + 1, v.y);
    atomicAdd(o + 2, v.z); atomicAdd(o + 3, v.w);
}

// 64-channel scatter: AGG[dst] += H[src]; thread owns (edge, 4-channel quad)
__global__ __launch_bounds__(256) void scatter64_kernel(
    const float* __restrict__ H, const int* __restrict__ src,
    const int* __restrict__ dst, float* __restrict__ agg) {
    long id = (long)blockIdx.x * 256 + threadIdx.x;  // E_TOT*16 threads
    long e = id >> 4;
    int  q = (int)(id & 15);
    int s = src[e], d = dst[e];
    float4 v = ((const float4*)(H + (size_t)s * HID))[q];
    float* o = agg + (size_t)d * HID + q * 4;
    atomicAdd(o + 0, v.x); atomicAdd(o + 1, v.y);
    atomicAdd(o + 2, v.z); atomicAdd(o + 3, v.w);
}

// ---------------- fused GIN MLP:  u = relu( relu(h@Wa+ba) @ Wb + bb ), + BN stats ----------------
// 128 threads = 4 waves; each wave computes a 16-node x 64-channel tile with
// V_WMMA_F32_16X16X4_F32 (fp32-exact).  K_IN = 4 (layer 1) or 64 (layers 2/3).
template <int K_IN>
__global__ __launch_bounds__(128) void gin_mlp_kernel(
    const float* __restrict__ hin,                         // N x K_IN
    const float* __restrict__ Wa, const float* __restrict__ ba,   // K_IN x 64, 64
    const float* __restrict__ Wb, const float* __restrict__ bb,   // 64 x 64, 64
    float* __restrict__ uout,                              // N x 64 (post-ReLU)
    float* __restrict__ gstats) {                          // [0:64)=sum, [64:128)=sumsq
    constexpr int TS = 66;                 // LDS row stride (floats): bank-conflict pad, 8B aligned
    __shared__ float sWa[(K_IN / 2) * HID * 2];   // (k-pair, n) float2 swizzle for B fragments
    __shared__ float sWb[(HID / 2) * HID * 2];
    __shared__ float sTile[4][16 * TS];
    __shared__ float sStats[128];

    const int tid  = threadIdx.x;
    const int wid  = tid >> 5;
    const int lane = tid & 31;
    const int ln   = lane & 15;            // column / row-in-halfwave index
    const int kh   = lane >> 4;            // half-wave: selects K pair per ISA fragment layout

    // stage weights, pre-swizzled: sW[(k>>1)*64 + n] = {W[k&~1][n], W[k|1][n]}
    for (int i = tid; i < K_IN * HID; i += 128) {
        int k = i >> 6, n = i & 63;
        sWa[((k >> 1) * HID + n) * 2 + (k & 1)] = Wa[i];
    }
    for (int i = tid; i < HID * HID; i += 128) {
        int k = i >> 6, n = i & 63;
        sWb[((k >> 1) * HID + n) * 2 + (k & 1)] = Wb[i];
    }
    if (tid < 128) sStats[tid] = 0.0f;
    __syncthreads();

    float* tile = sTile[wid];
    const long node0 = ((long)blockIdx.x * 4 + wid) * 16;

    // stage 16 x K_IN input tile (row-major, stride TS)
    {
        const float* srcp = hin + node0 * K_IN;
        constexpr int C4 = K_IN / 4;
        for (int i = lane; i < 16 * C4; i += 32) {
            int r = i / C4, c = i % C4;
            float4 v = ((const float4*)srcp)[i];
            float* t = &tile[r * TS + c * 4];
            ((v2f*)t)[0] = (v2f){v.x, v.y};
            ((v2f*)t)[1] = (v2f){v.z, v.w};
        }
    }
    __syncthreads();

    // ---- GEMM1: t = relu(h @ Wa + ba), 16x64 output as 4 n-tiles ----
    v8f acc[4];
#pragma unroll
    for (int nt = 0; nt < 4; ++nt) acc[nt] = splat8(ba[nt * 16 + ln]);
#pragma unroll
    for (int kk = 0; kk < K_IN / 4; ++kk) {
        // A fragment (16x4 f32): VGPR0={K=0|2}, VGPR1={K=1|3} per half-wave
        v2f a = *(const v2f*)&tile[ln * TS + kk * 4 + kh * 2];
        int p = kk * 2 + kh;
#pragma unroll
        for (int nt = 0; nt < 4; ++nt) {
            v2f b = *(const v2f*)&sWa[(p * HID + nt * 16 + ln) * 2];
            acc[nt] = __builtin_amdgcn_wmma_f32_16x16x4_f32(
                false, a, false, b, (short)0, acc[nt], false, false);
        }
    }
    __syncthreads();   // all fragment reads of 'tile' done before overwrite

    // transpose-through-LDS: write t (C/D layout -> row-major A layout), with ReLU
#pragma unroll
    for (int nt = 0; nt < 4; ++nt) {
#pragma unroll
        for (int v = 0; v < 8; ++v) {
            float x = acc[nt][v];
            x = x > 0.0f ? x : 0.0f;
            tile[(v + kh * 8) * TS + nt * 16 + ln] = x;
        }
    }
    __syncthreads();

    // ---- GEMM2: u = relu(t @ Wb + bb) ----
    v8f acc2[4];
#pragma unroll
    for (int nt = 0; nt < 4; ++nt) acc2[nt] = splat8(bb[nt * 16 + ln]);
#pragma unroll
    for (int kk = 0; kk < HID / 4; ++kk) {
        v2f a = *(const v2f*)&tile[ln * TS + kk * 4 + kh * 2];
        int p = kk * 2 + kh;
#pragma unroll
        for (int nt = 0; nt < 4; ++nt) {
            v2f b = *(const v2f*)&sWb[(p * HID + nt * 16 + ln) * 2];
            acc2[nt] = __builtin_amdgcn_wmma_f32_16x16x4_f32(
                false, a, false, b, (short)0, acc2[nt], false, false);
        }
    }

    // ReLU + store + per-channel BN statistics
    float* outp = uout + node0 * HID;
#pragma unroll
    for (int nt = 0; nt < 4; ++nt) {
        float s = 0.0f, ss = 0.0f;
#pragma unroll
        for (int v = 0; v < 8; ++v) {
            float x = acc2[nt][v];
            x = x > 0.0f ? x : 0.0f;
            outp[(v + kh * 8) * HID + nt * 16 + ln] = x;
            s += x; ss += x * x;
        }
        atomicAdd(&sStats[nt * 16 + ln], s);
        atomicAdd(&sStats[64 + nt * 16 + ln], ss);
    }
    __syncthreads();
    if (tid < 128) atomicAdd(&gstats[tid], sStats[tid]);
}

// ---------------- BN finalize: fold stats+gamma/beta into scale/shift ----------------
__global__ void bn_finalize_kernel(const float* __restrict__ stats,
                                   const float* __restrict__ g,
                                   const float* __restrict__ be,
                                   float* __restrict__ bnp) {
    int c = threadIdx.x;
    if (c < HID) {
        const float inv = 1.0f / (float)N_TOT;
        float m  = stats[c] * inv;
        float vv = stats[64 + c] * inv - m * m;
        float sc = g[c] * rsqrtf(vv + BN_EPS);
        bnp[c]      = sc;
        bnp[64 + c] = be[c] - m * sc;
    }
}

// ---------------- BN apply (in place) + optional self-term copy into AGG ----------------
__global__ __launch_bounds__(256) void bn_apply_kernel(
    float* __restrict__ U, const float* __restrict__ bnp, float* __restrict__ aggcopy) {
    __shared__ float s[128];
    if (threadIdx.x < 128) s[threadIdx.x] = bnp[threadIdx.x];
    __syncthreads();
    long id = (long)blockIdx.x * 256 + threadIdx.x;   // N_TOT*16 threads (float4 each)
    int c = (int)(id & 15) * 4;
    float4 v = ((const float4*)U)[id];
    v.x = v.x * s[c + 0] + s[64 + c + 0];
    v.y = v.y * s[c + 1] + s[64 + c + 1];
    v.z = v.z * s[c + 2] + s[64 + c + 2];
    v.w = v.w * s[c + 3] + s[64 + c + 3];
    ((float4*)U)[id] = v;
    if (aggcopy) ((float4*)aggcopy)[id] = v;
}

// ---------------- per-graph sum pool + 64->3 FC ----------------
__global__ __launch_bounds__(256) void pool_fc_kernel(
    const float* __restrict__ H, const float* __restrict__ Wfc,
    const float* __restrict__ bfc, float* __restrict__ out) {
    int wid = threadIdx.x >> 5, lane = threadIdx.x & 31;
    int g = blockIdx.x * 8 + wid;                  // one wave per graph
    const float* base = H + (size_t)g * NODES_PG * HID;
    float a0 = 0.0f, a1 = 0.0f;
    for (int r = 0; r < NODES_PG; ++r) {
        a0 += base[r * HID + lane];
        a1 += base[r * HID + 32 + lane];
    }
#pragma unroll
    for (int k = 0; k < 3; ++k) {
        float p = a0 * Wfc[lane * 3 + k] + a1 * Wfc[(lane + 32) * 3 + k];
#pragma unroll
        for (int off = 16; off > 0; off >>= 1) p += __shfl_down(p, off, 32);
        if (lane == 0) out[g * 3 + k] = p + bfc[k];
    }
}

// ---------------- host orchestration ----------------
extern "C" void kernel_launch(void* const* d_in, const int* in_sizes, int n_in,
                              void* d_out, int out_size, void* d_ws, size_t ws_size,
                              hipStream_t stream) {
    const float* x    = (const float*)d_in[0];
    const int*   esrc = (const int*)d_in[1];
    const int*   edst = (const int*)d_in[2];
    // d_in[3] = num_graphs (compile-time constant here)
    const float *W1a = (const float*)d_in[4],  *b1a = (const float*)d_in[5];
    const float *W1b = (const float*)d_in[6],  *b1b = (const float*)d_in[7];
    const float *g1  = (const float*)d_in[8],  *be1 = (const float*)d_in[9];
    const float *W2a = (const float*)d_in[10], *b2a = (const float*)d_in[11];
    const float *W2b = (const float*)d_in[12], *b2b = (const float*)d_in[13];
    const float *g2  = (const float*)d_in[14], *be2 = (const float*)d_in[15];
    const float *W3a = (const float*)d_in[16], *b3a = (const float*)d_in[17];
    const float *W3b = (const float*)d_in[18], *b3b = (const float*)d_in[19];
    const float *g3  = (const float*)d_in[20], *be3 = (const float*)d_in[21];
    const float *Wfc = (const float*)d_in[22], *bfc = (const float*)d_in[23];
    float* out = (float*)d_out;

    // workspace layout
    float* U     = (float*)d_ws;                  // N x 64
    float* AGG   = U   + (size_t)N_TOT * HID;     // N x 64
    float* AGG1  = AGG + (size_t)N_TOT * HID;     // N x 4
    float* STATS = AGG1 + (size_t)N_TOT * 4;      // 128
    float* BNP   = STATS + 128;                   // 128

    const int GRID_MLP   = N_TOT / 64;            // 4 waves x 16 nodes per block
    const int GRID_BN    = (N_TOT * 16) / 256;
    const int GRID_SC64  = (int)(((long)E_TOT * 16) / 256);

    // ---- layer 1 ----
    init_copy4_kernel<<<N_TOT / 256, 256, 0, stream>>>((const float4*)x, (float4*)AGG1);
    scatter4_kernel<<<E_TOT / 256, 256, 0, stream>>>(x, esrc, edst, AGG1);
    hipMemsetAsync(STATS, 0, 128 * sizeof(float), stream);
    gin_mlp_kernel<4><<<GRID_MLP, 128, 0, stream>>>(AGG1, W1a, b1a, W1b, b1b, U, STATS);
    bn_finalize_kernel<<<1, 64, 0, stream>>>(STATS, g1, be1, BNP);

    // ---- layer 2 ----
    bn_apply_kernel<<<GRID_BN, 256, 0, stream>>>(U, BNP, AGG);
    scatter64_kernel<<<GRID_SC64, 256, 0, stream>>>(U, esrc, edst, AGG);
    hipMemsetAsync(STATS, 0, 128 * sizeof(float), stream);
    gin_mlp_kernel<64><<<GRID_MLP, 128, 0, stream>>>(AGG, W2a, b2a, W2b, b2b, U, STATS);
    bn_finalize_kernel<<<1, 64, 0, stream>>>(STATS, g2, be2, BNP);

    // ---- layer 3 ----
    bn_apply_kernel<<<GRID_BN, 256, 0, stream>>>(U, BNP, AGG);
    scatter64_kernel<<<GRID_SC64, 256, 0, stream>>>(U, esrc, edst, AGG);
    hipMemsetAsync(STATS, 0, 128 * sizeof(float), stream);
    gin_mlp_kernel<64><<<GRID_MLP, 128, 0, stream>>>(AGG, W3a, b3a, W3b, b3b, U, STATS);
    bn_finalize_kernel<<<1, 64, 0, stream>>>(STATS, g3, be3, BNP);

    // ---- final BN apply + pool + FC ----
    bn_apply_kernel<<<GRID_BN, 256, 0, stream>>>(U, BNP, (float*)nullptr);
    pool_fc_kernel<<<N_GRAPHS / 8, 256, 0, stream>>>(U, Wfc, bfc, out);
}